// Bispectrum_2843268350801
// MI455X (gfx1250) — compile-verified
//
#include <hip/hip_runtime.h>
#include <hip/hip_bf16.h>

#define NATOM 20000
#define NEIGH 600000
#define NWAVE 8
#define NSPH  16

typedef float v2f __attribute__((ext_vector_type(2)));
typedef float v8f __attribute__((ext_vector_type(8)));

__device__ __forceinline__ float silu(float x) { return x / (1.0f + expf(-x)); }

// ---------------------------------------------------------------------------
// Kernel 1: per-atom embedding MLP, 16 atoms per wave, fp32 WMMA 16x16x4.
//   h1 = silu(species*W1 + b1)          (rank-1, VALU)
//   h2 = silu(h1 @ W2 + b2)             (4x  v_wmma_f32_16x16x4_f32)
//   emb = h2 @ W3 + b3  (24 cols)       (8x  v_wmma_f32_16x16x4_f32, N split 16+8)
// A-layout (fp32 16x4): lane L holds M=L&15, K = k0 + (L<16 ? {0,1} : {2,3})
// B-layout (fp32 4x16): VGPR v holds row K = k0 + (L<16 ? v : v+2), col N=L&15
// C-layout (fp32 16x16): VGPR r holds M = r + (L<16?0:8), N = L&15
// ---------------------------------------------------------------------------
__global__ __launch_bounds__(32)
void emb_kernel(const float* __restrict__ species,
                const float* __restrict__ W1, const float* __restrict__ b1,
                const float* __restrict__ W2, const float* __restrict__ b2,
                const float* __restrict__ W3, const float* __restrict__ b3,
                float* __restrict__ emb)
{
    __shared__ float hbuf[16][17];          // 16x16 activation tile, padded rows
    const int lane  = threadIdx.x;
    const int atom0 = blockIdx.x * 16;
    const int n     = lane & 15;
    const bool hi   = lane >= 16;

    // ---- layer 1: h1[m][f] = silu(species[m]*W1[f] + b1[f]) ----
    if (!hi) {
        float sp = species[atom0 + lane];
        #pragma unroll
        for (int f = 0; f < 16; ++f)
            hbuf[lane][f] = silu(fmaf(sp, W1[f], b1[f]));
    }
    __syncthreads();

    // ---- layer 2: h2 = silu(h1 @ W2 + b2), K=16 as 4x WMMA k-steps ----
    v8f c;
    #pragma unroll
    for (int r = 0; r < 8; ++r) c[r] = b2[n];
    #pragma unroll
    for (int kb = 0; kb < 4; ++kb) {
        const int k0 = kb * 4 + (hi ? 2 : 0);
        v2f a, b;
        a.x = hbuf[n][k0];
        a.y = hbuf[n][k0 + 1];
        b.x = W2[k0 * 16 + n];
        b.y = W2[(k0 + 1) * 16 + n];
        c = __builtin_amdgcn_wmma_f32_16x16x4_f32(false, a, false, b,
                                                  (short)0, c, false, false);
    }
    __syncthreads();
    #pragma unroll
    for (int r = 0; r < 8; ++r) {
        const int m = r + (hi ? 8 : 0);
        hbuf[m][n] = silu(c[r]);
    }
    __syncthreads();

    // ---- layer 3: emb = h2 @ W3 + b3, W3 is 16x24 -> N blocks [0..15], [16..23] ----
    v8f c0, c1;
    #pragma unroll
    for (int r = 0; r < 8; ++r) {
        c0[r] = b3[n];
        c1[r] = (n < 8) ? b3[16 + n] : 0.0f;
    }
    #pragma unroll
    for (int kb = 0; kb < 4; ++kb) {
        const int k0 = kb * 4 + (hi ? 2 : 0);
        v2f a, bA, bB;
        a.x = hbuf[n][k0];
        a.y = hbuf[n][k0 + 1];
        bA.x = W3[k0 * 24 + n];
        bA.y = W3[(k0 + 1) * 24 + n];
        bB.x = (n < 8) ? W3[k0 * 24 + 16 + n]       : 0.0f;
        bB.y = (n < 8) ? W3[(k0 + 1) * 24 + 16 + n] : 0.0f;
        c0 = __builtin_amdgcn_wmma_f32_16x16x4_f32(false, a, false, bA,
                                                   (short)0, c0, false, false);
        c1 = __builtin_amdgcn_wmma_f32_16x16x4_f32(false, a, false, bB,
                                                   (short)0, c1, false, false);
    }
    #pragma unroll
    for (int r = 0; r < 8; ++r) {
        const int m = r + (hi ? 8 : 0);
        emb[(atom0 + m) * 24 + n] = c0[r];
        if (n < 8) emb[(atom0 + m) * 24 + 16 + n] = c1[r];
    }
}

// ---------------------------------------------------------------------------
// Kernel 2: per-neighbor radial x angular outer product, scattered into
// sum_sph[atom][sph(16)][wave(8)] with L2-resident f32 atomics.
// ---------------------------------------------------------------------------
__global__ __launch_bounds__(256)
void neigh_kernel(const float* __restrict__ cart,
                  const int*   __restrict__ aidx,
                  const float* __restrict__ shifts,
                  const float* __restrict__ emb,
                  float* __restrict__ sum_sph)
{
    int nn = blockIdx.x * blockDim.x + threadIdx.x;
    if (nn >= NEIGH) return;
    const int i = aidx[nn];
    const int j = aidx[NEIGH + nn];

    float dx = cart[j]           - cart[i]           + shifts[nn];
    float dy = cart[NATOM + j]   - cart[NATOM + i]   + shifts[NEIGH + nn];
    float dz = cart[2*NATOM + j] - cart[2*NATOM + i] + shifts[2*NEIGH + nn];
    float dist = sqrtf(dx*dx + dy*dy + dz*dz);
    float inv  = 1.0f / dist;
    float x = dx * inv, y = dy * inv, z = dz * inv;

    // cutoff ((cos(pi*d/5)+1)/2)^3
    float cc  = (cosf(dist * 0.6283185307179586f) + 1.0f) * 0.5f;
    float cut = cc * cc * cc;

    // radial: cut * coeff_w * exp(-(alpha_w*(d-center_w))^2)
    const float* e = emb + j * 24;
    float fw[NWAVE];
    #pragma unroll
    for (int w = 0; w < NWAVE; ++w) {
        float sd = e[8 + w] * (dist - e[16 + w]);
        fw[w] = cut * e[w] * expf(-sd * sd);
    }

    // real spherical harmonics (l<=3), reference convention (CS phase in P)
    float s2c = 1.0f - z * z; if (s2c < 0.0f) s2c = 0.0f;
    float s   = sqrtf(s2c);
    float phi = atan2f(y, x);
    float sp1, cp1; sincosf(phi, &sp1, &cp1);
    float cp2 = cp1*cp1 - sp1*sp1, sp2 = 2.0f*sp1*cp1;
    float cp3 = cp1*cp2 - sp1*sp2, sp3 = sp1*cp2 + cp1*sp2;

    float P11 = -s;
    float P22 = -3.0f * s * P11;
    float P33 = -5.0f * s * P22;
    float P10 = z;
    float P21 = 3.0f * z * P11;
    float P32 = 5.0f * z * P22;
    float P20 = 0.5f * (3.0f * z * P10 - 1.0f);
    float P30 = (5.0f * z * P20 - 2.0f * P10) * (1.0f / 3.0f);
    float P31 = 0.5f * (5.0f * z * P21 - 3.0f * P11);

    float sph[NSPH];
    sph[0]  = 0.28209479177387814f;
    sph[1]  = 0.4886025119029199f * sp1 * P11;
    sph[2]  = 0.4886025119029199f * P10;
    sph[3]  = 0.4886025119029199f * cp1 * P11;
    sph[4]  = 0.18209140509868f   * sp2 * P22;
    sph[5]  = 0.36418281019737f   * sp1 * P21;
    sph[6]  = 0.63078313050504f   * P20;
    sph[7]  = 0.36418281019737f   * cp1 * P21;
    sph[8]  = 0.18209140509868f   * cp2 * P22;
    sph[9]  = 0.039335678393723f  * sp3 * P33;
    sph[10] = 0.096350591694069f  * sp2 * P32;
    sph[11] = 0.30469719964297f   * sp1 * P31;
    sph[12] = 0.74635266518023f   * P30;
    sph[13] = 0.30469719964297f   * cp1 * P31;
    sph[14] = 0.096350591694069f  * cp2 * P32;
    sph[15] = 0.039335678393723f  * cp3 * P33;

    float* base = sum_sph + (size_t)i * (NSPH * NWAVE);
    #pragma unroll
    for (int k = 0; k < NSPH; ++k) {
        #pragma unroll
        for (int w = 0; w < NWAVE; ++w)
            unsafeAtomicAdd(base + k * NWAVE + w, sph[k] * fw[w]);
    }
}

// ---------------------------------------------------------------------------
// Kernel 3: Wigner-3j contraction. 80 terms, constants verified against
// the orthogonality sum rule per (J, m3) block.
// ---------------------------------------------------------------------------
struct Term { unsigned char a, b, c, j; float w; };
__device__ constexpr Term TERMS[80] = {
    // J=[1,1,0]
    {1,3,0,0,  0.57735027f}, {2,2,0,0, -0.57735027f}, {3,1,0,0,  0.57735027f},
    // J=[1,1,2]
    {1,1,8,1,  0.44721360f}, {1,2,7,1, -0.31622777f}, {1,3,6,1,  0.18257419f},
    {2,1,7,1, -0.31622777f}, {2,2,6,1,  0.36514837f}, {2,3,5,1, -0.31622777f},
    {3,1,6,1,  0.18257419f}, {3,2,5,1, -0.31622777f}, {3,3,4,1,  0.44721360f},
    // J=[2,2,2]
    {4,6,8,2,  0.23904572f}, {4,7,7,2, -0.29277002f}, {4,8,6,2,  0.23904572f},
    {5,5,8,2, -0.29277002f}, {5,6,7,2,  0.11952286f}, {5,7,6,2,  0.11952286f},
    {5,8,5,2, -0.29277002f}, {6,4,8,2,  0.23904572f}, {6,5,7,2,  0.11952286f},
    {6,6,6,2, -0.23904572f}, {6,7,5,2,  0.11952286f}, {6,8,4,2,  0.23904572f},
    {7,4,7,2, -0.29277002f}, {7,5,6,2,  0.11952286f}, {7,6,5,2,  0.11952286f},
    {7,7,4,2, -0.29277002f}, {8,4,6,2,  0.23904572f}, {8,5,5,2, -0.29277002f},
    {8,6,4,2,  0.23904572f},
    // J=[2,2,0]
    {4,8,0,3,  0.44721360f}, {5,7,0,3, -0.44721360f}, {6,6,0,3,  0.44721360f},
    {7,5,0,3, -0.44721360f}, {8,4,0,3,  0.44721360f},
    // J=[1,2,3]
    {1,4,15,4,  0.37796447f}, {1,5,14,4, -0.30860670f}, {1,6,13,4,  0.23904572f},
    {1,7,12,4, -0.16903085f}, {1,8,11,4,  0.09759001f},
    {2,4,14,4, -0.21821789f}, {2,5,13,4,  0.27602622f}, {2,6,12,4, -0.29277002f},
    {2,7,11,4,  0.27602622f}, {2,8,10,4, -0.21821789f},
    {3,4,13,4,  0.09759001f}, {3,5,12,4, -0.16903085f}, {3,6,11,4,  0.23904572f},
    {3,7,10,4, -0.30860670f}, {3,8, 9,4,  0.37796447f},
    // J=[3,3,2]
    { 9,13,8,5,  0.15430335f}, { 9,14,7,5, -0.24397502f}, { 9,15,6,5,  0.24397502f},
    {10,12,8,5, -0.21821789f}, {10,13,7,5,  0.18898224f}, {10,14,6,5,  0.0f},
    {10,15,5,5, -0.24397502f},
    {11,11,8,5,  0.23904572f}, {11,12,7,5, -0.06900656f}, {11,13,6,5, -0.14638501f},
    {11,14,5,5,  0.18898224f}, {11,15,4,5,  0.15430335f},
    {12,10,8,5, -0.21821789f}, {12,11,7,5, -0.06900656f}, {12,12,6,5,  0.19518001f},
    {12,13,5,5, -0.06900656f}, {12,14,4,5, -0.21821789f},
    {13, 9,8,5,  0.15430335f}, {13,10,7,5,  0.18898224f}, {13,11,6,5, -0.14638501f},
    {13,12,5,5, -0.06900656f}, {13,13,4,5,  0.23904572f},
    {14, 9,7,5, -0.24397502f}, {14,10,6,5,  0.0f},        {14,11,5,5,  0.18898224f},
    {14,12,4,5, -0.21821789f},
    {15, 9,6,5,  0.24397502f}, {15,10,5,5, -0.24397502f}, {15,11,4,5,  0.15430335f},
};

__global__ __launch_bounds__(256)
void dens_kernel(const float* __restrict__ sum_sph, float* __restrict__ out)
{
    int t = blockIdx.x * blockDim.x + threadIdx.x;   // t = atom*8 + wave
    if (t >= NATOM * NWAVE) return;
    const int a = t >> 3, w = t & 7;

    const float* sp = sum_sph + (size_t)a * (NSPH * NWAVE) + w;
    float S[NSPH];
    #pragma unroll
    for (int k = 0; k < NSPH; ++k) S[k] = sp[k * NWAVE];

    float acc[6] = {0.f, 0.f, 0.f, 0.f, 0.f, 0.f};
    #pragma unroll
    for (int q = 0; q < 80; ++q)
        acc[TERMS[q].j] = fmaf(TERMS[q].w,
                               S[TERMS[q].a] * S[TERMS[q].b] * S[TERMS[q].c],
                               acc[TERMS[q].j]);

    #pragma unroll
    for (int q = 0; q < 6; ++q)
        out[(size_t)q * NATOM * NWAVE + t] = acc[q];
}

// ---------------------------------------------------------------------------
extern "C" void kernel_launch(void* const* d_in, const int* in_sizes, int n_in,
                              void* d_out, int out_size, void* d_ws, size_t ws_size,
                              hipStream_t stream)
{
    const float* cart    = (const float*)d_in[0];
    const int*   aidx    = (const int*)  d_in[1];
    const float* shifts  = (const float*)d_in[2];
    const float* species = (const float*)d_in[3];
    const float* W1      = (const float*)d_in[4];
    const float* b1      = (const float*)d_in[5];
    const float* W2      = (const float*)d_in[6];
    const float* b2      = (const float*)d_in[7];
    const float* W3      = (const float*)d_in[8];
    const float* b3      = (const float*)d_in[9];
    float* out = (float*)d_out;

    float* emb     = (float*)d_ws;                    // NATOM * 24 floats
    float* sum_sph = emb + (size_t)NATOM * 24;        // NATOM * 16 * 8 floats

    hipMemsetAsync(sum_sph, 0, (size_t)NATOM * NSPH * NWAVE * sizeof(float), stream);

    emb_kernel<<<NATOM / 16, 32, 0, stream>>>(species, W1, b1, W2, b2, W3, b3, emb);
    neigh_kernel<<<(NEIGH + 255) / 256, 256, 0, stream>>>(cart, aidx, shifts, emb, sum_sph);
    dens_kernel<<<(NATOM * NWAVE + 255) / 256, 256, 0, stream>>>(sum_sph, out);
}